// SFMCNN_89824946029330
// MI455X (gfx1250) — compile-verified
//
#include <hip/hip_runtime.h>
#include <hip/hip_bf16.h>

typedef __attribute__((ext_vector_type(2))) float v2f;
typedef __attribute__((ext_vector_type(8))) float v8f;

#define KCHUNK 32
#define LDSTRIDE 38   // even (8B-aligned b64 reads), gcd(38,64)=2 -> 32 distinct banks

// ---------------------------------------------------------------------------
// Implicit-GEMM RBF distance kernel using V_WMMA_F32_16X16X4_F32.
// One wave computes a 32(M) x 32(N) tile with 4 accumulators:
//   [a0;a1] x [b0,b1] -> acc00 acc01 acc10 acc11   (4 WMMA per k-step of 4)
// A is gathered implicitly (im2col on the fly) into LDS once per K chunk;
// the k -> (c,ky,kx) decomposition is done ONCE per lane per chunk and
// broadcast via readlane (SALU), so the gather loop has no div/mod.
// Epilogue fuses d = sqrt(max(pnorm + wnorm - 2*dot, 1e-12)).
// Output layout: dout[b, n, p] (channel-major for the top-k stage).
// ---------------------------------------------------------------------------
__global__ __launch_bounds__(32) void rbf_gemm_wmma(
    const float* __restrict__ in, long in_bstride, long in_cstride,
    int Cin, int H, int W, int Kk, int pad, int Kdim,
    const float* __restrict__ Wt,     // [Cout, Kdim] (branch slice)
    const float* __restrict__ pnorm,  // [M]
    const float* __restrict__ wnorm,  // [Cout]
    float* __restrict__ dout,         // [B, Cout, H*W]
    int Cout)
{
    __shared__ __align__(16) float lds_a[32][LDSTRIDE];  // [row][k]
    __shared__ __align__(16) float lds_b[32][LDSTRIDE];  // [col][k]

    const int lane = threadIdx.x;            // 0..31
    const int m0 = blockIdx.x * 32;
    const int n0 = blockIdx.y * 32;
    const int HW = H * W;
    const int KK2 = Kk * Kk;
    const int klo = (lane >> 4) * 2;         // A/B fragment k sub-offset: 0 or 2
    const int hl  = lane & 15;

    // ---- per-lane row (each lane owns one of the 32 M rows for the gather) ----
    const int mrow = m0 + lane;
    const int ab = mrow / HW;
    const int ap = mrow - ab * HW;
    const int ay = ap / W;
    const int ax = ap - ay * W;
    const float* abase = in + (long)ab * in_bstride + ap;  // + klin gives the element

    v8f acc00 = {}, acc01 = {}, acc10 = {}, acc11 = {};

    const int nchunks = (Kdim + KCHUNK - 1) / KCHUNK;
    for (int ch = 0; ch < nchunks; ++ch) {
        const int kbase = ch * KCHUNK;

        // ---- per-lane k-table entry for kk = kbase + lane (one div/mod per chunk) ----
        const int kk = kbase + lane;
        int c  = kk / KK2;
        int r  = kk - c * KK2;
        int ky = r / Kk;
        int kx = r - ky * Kk;
        int dy = ky - pad;
        int dx = kx - pad;
        int klin = c * (int)in_cstride + dy * W + dx;
        int kval = (kk < Kdim) ? 1 : 0;

        // prefetch next weight chunk (gfx1250 global_prefetch)
        if (kbase + KCHUNK < Kdim)
            __builtin_prefetch(Wt + (long)(n0 + lane) * Kdim + kbase + KCHUNK, 0, 1);

        // ---- stage B: lds_b[col][k], coalesced along k ----
#pragma unroll 4
        for (int i = 0; i < 32; ++i) {
            float v = kval ? Wt[(long)(n0 + i) * Kdim + kk] : 0.0f;
            lds_b[i][lane] = v;
        }

        // ---- stage A: gather patch elements; lanes sweep rows (coalesced) ----
#pragma unroll 4
        for (int i = 0; i < 32; ++i) {
            int dyi = __builtin_amdgcn_readlane(dy, i);
            int dxi = __builtin_amdgcn_readlane(dx, i);
            int kli = __builtin_amdgcn_readlane(klin, i);
            int kvi = __builtin_amdgcn_readlane(kval, i);
            int iy = ay + dyi;
            int ix = ax + dxi;
            float v = 0.0f;
            if (kvi && iy >= 0 && iy < H && ix >= 0 && ix < W)
                v = abase[kli];
            lds_a[lane][i] = v;
        }
        __syncthreads();

        // ---- MMA: 8 k-steps x 4 WMMA, fragments from LDS (ds_load_b64) ----
#pragma unroll
        for (int ks = 0; ks < KCHUNK; ks += 4) {
            v2f a0, a1, b0, b1;
            a0.x = lds_a[hl +  0][ks + klo + 0];
            a0.y = lds_a[hl +  0][ks + klo + 1];
            a1.x = lds_a[hl + 16][ks + klo + 0];
            a1.y = lds_a[hl + 16][ks + klo + 1];
            b0.x = lds_b[hl +  0][ks + klo + 0];
            b0.y = lds_b[hl +  0][ks + klo + 1];
            b1.x = lds_b[hl + 16][ks + klo + 0];
            b1.y = lds_b[hl + 16][ks + klo + 1];
            acc00 = __builtin_amdgcn_wmma_f32_16x16x4_f32(false, a0, false, b0, (short)0, acc00, false, false);
            acc01 = __builtin_amdgcn_wmma_f32_16x16x4_f32(false, a0, false, b1, (short)0, acc01, false, false);
            acc10 = __builtin_amdgcn_wmma_f32_16x16x4_f32(false, a1, false, b0, (short)0, acc10, false, false);
            acc11 = __builtin_amdgcn_wmma_f32_16x16x4_f32(false, a1, false, b1, (short)0, acc11, false, false);
        }
        __syncthreads();
    }

    // ---- epilogue: d = sqrt(max(||p||^2 + ||w||^2 - 2*dot, 1e-12)) ----
    const float wn0 = wnorm[n0 + hl];
    const float wn1 = wnorm[n0 + 16 + hl];
#pragma unroll
    for (int mh = 0; mh < 2; ++mh) {
#pragma unroll
        for (int j = 0; j < 8; ++j) {
            int row = m0 + mh * 16 + j + ((lane >> 4) << 3);  // lanes 0-15: M=j, 16-31: M=8+j
            int b = row / HW;
            int p = row - b * HW;
            float pn = pnorm[row];
            float v0 = (mh == 0) ? acc00[j] : acc10[j];
            float v1 = (mh == 0) ? acc01[j] : acc11[j];
            float d0 = sqrtf(fmaxf(pn + wn0 - 2.0f * v0, 1e-12f));
            float d1 = sqrtf(fmaxf(pn + wn1 - 2.0f * v1, 1e-12f));
            dout[((long)b * Cout + (n0 + hl)) * HW + p] = d0;
            dout[((long)b * Cout + (n0 + 16 + hl)) * HW + p] = d1;
        }
    }
}

// ---------------------------------------------------------------------------
// pnorm[b,y,x] = sum over patch of in^2 (box filter of squares, zero-padded)
// ---------------------------------------------------------------------------
__global__ __launch_bounds__(256) void pnorm_kernel(
    const float* __restrict__ in, long bstride, long cstride,
    int Cin, int H, int W, int Kk, int pad,
    float* __restrict__ pn, long M)
{
    long m = (long)blockIdx.x * blockDim.x + threadIdx.x;
    if (m >= M) return;
    int HW = H * W;
    int b = (int)(m / HW);
    int p = (int)(m - (long)b * HW);
    int y = p / W;
    int x = p - y * W;
    const float* base = in + (long)b * bstride;
    float s = 0.0f;
    for (int c = 0; c < Cin; ++c) {
        const float* cb = base + (long)c * cstride;
        for (int ky = 0; ky < Kk; ++ky) {
            int iy = y + ky - pad;
            if (iy < 0 || iy >= H) continue;
            for (int kx = 0; kx < Kk; ++kx) {
                int ix = x + kx - pad;
                if (ix < 0 || ix >= W) continue;
                float v = cb[(long)iy * W + ix];
                s += v * v;
            }
        }
    }
    pn[m] = s;
}

// wnorm[o] = sum_k W[o,k]^2
__global__ void wnorm_kernel(const float* __restrict__ Wt, int Kdim,
                             float* __restrict__ wn, int Cout)
{
    int o = blockIdx.x * blockDim.x + threadIdx.x;
    if (o >= Cout) return;
    const float* w = Wt + (long)o * Kdim;
    float s = 0.0f;
    for (int k = 0; k < Kdim; ++k) { float v = w[k]; s += v * v; }
    wn[o] = s;
}

// ---------------------------------------------------------------------------
// triangle_crelu: one wave per pixel; channels in registers (VPL per lane).
// Exact k-th largest over C positive floats via 31-step bisection on the
// monotone IEEE bit pattern, wave-reduced counts via __shfl_xor.
// out = (d > min(thr, w)) ? 0 : 1 - d/w
// ---------------------------------------------------------------------------
template <int VPL>
__global__ __launch_bounds__(256) void crelu_kernel(
    const float* __restrict__ din,   // [B, C, HW]
    float* __restrict__ dout,        // out[b*ob + c*oc + p]
    long out_bstride, long out_cstride,
    int C, int HW, int B, int kth,
    const float* __restrict__ wvp)
{
    const int wid  = threadIdx.x >> 5;
    const int lane = threadIdx.x & 31;
    long pix = (long)blockIdx.x * 8 + wid;
    long npix = (long)B * HW;
    if (pix >= npix) return;
    int b = (int)(pix / HW);
    int p = (int)(pix - (long)b * HW);
    const float* base = din + ((long)b * C) * HW + p;

    float vals[VPL];
#pragma unroll
    for (int j = 0; j < VPL; ++j)
        vals[j] = base[(long)(lane + 32 * j) * HW];

    // bisection on bits: all vals >= sqrt(1e-12) > 0, so uint order == float order
    unsigned lo = 0u, hi = 0x7f800000u;
    for (int it = 0; it < 31; ++it) {
        unsigned mid = (lo + hi) >> 1;
        int cnt = 0;
#pragma unroll
        for (int j = 0; j < VPL; ++j)
            cnt += (__float_as_uint(vals[j]) >= mid) ? 1 : 0;
#pragma unroll
        for (int off = 16; off > 0; off >>= 1)
            cnt += __shfl_xor(cnt, off, 32);
        if (cnt >= kth) lo = mid; else hi = mid;
    }
    const float wv  = wvp[0];
    const float thr = fminf(__uint_as_float(lo), wv);
    const float inv_w = 1.0f / wv;
#pragma unroll
    for (int j = 0; j < VPL; ++j) {
        int c = lane + 32 * j;
        float v = vals[j];
        float o = (v > thr) ? 0.0f : (1.0f - v * inv_w);
        dout[(long)b * out_bstride + (long)c * out_cstride + p] = o;
    }
}

// ---------------------------------------------------------------------------
// sfm: 2x2 ALPHA-weighted mean. ALPHA = [[0.9,0.93],[0.96,0.99]], /4.
// ---------------------------------------------------------------------------
__global__ __launch_bounds__(256) void sfm_kernel(
    const float* __restrict__ in, float* __restrict__ out,
    int Hin, int Win, long total)
{
    long i = (long)blockIdx.x * blockDim.x + threadIdx.x;
    if (i >= total) return;
    int Wo = Win >> 1, Ho = Hin >> 1;
    int x = (int)(i % Wo);
    long t = i / Wo;
    int y = (int)(t % Ho);
    t /= Ho;                       // t = b*C + c
    const float* p = in + (t * Hin + 2L * y) * Win + 2L * x;
    float v = 0.90f * p[0] + 0.93f * p[1] + 0.96f * p[Win] + 0.99f * p[Win + 1];
    out[i] = 0.25f * v;
}

// ---------------------------------------------------------------------------
// Final FC: out[b, j] = feats[b,:] . fc_w[j,:] + fc_b[j]   (16 x 10)
// ---------------------------------------------------------------------------
__global__ __launch_bounds__(256) void fc_kernel(
    const float* __restrict__ feats, const float* __restrict__ fcw,
    const float* __restrict__ fcb, float* __restrict__ out, int fcin)
{
    __shared__ float red[256];
    const int b = blockIdx.x;
    const int j = blockIdx.y;
    const float* f = feats + (long)b * fcin;
    const float* w = fcw + (long)j * fcin;
    float s = 0.0f;
    for (int i = threadIdx.x; i < fcin; i += 256) s += f[i] * w[i];
    red[threadIdx.x] = s;
    __syncthreads();
    for (int st = 128; st > 0; st >>= 1) {
        if (threadIdx.x < st) red[threadIdx.x] += red[threadIdx.x + st];
        __syncthreads();
    }
    if (threadIdx.x == 0) out[b * 10 + j] = red[0] + fcb[j];
}

// ---------------------------------------------------------------------------
// Host orchestration
// ---------------------------------------------------------------------------
extern "C" void kernel_launch(void* const* d_in, const int* in_sizes, int n_in,
                              void* d_out, int out_size, void* d_ws, size_t ws_size,
                              hipStream_t stream)
{
    const float* x   = (const float*)d_in[0];   // (16,3,96,96)
    const float* w1  = (const float*)d_in[1];   // (3,64,25)
    const float* w2  = (const float*)d_in[2];   // (3,128,576)
    const float* w3  = (const float*)d_in[3];   // (3,256,1152)
    const float* tw  = (const float*)d_in[4];   // (3,3)
    const float* fcw = (const float*)d_in[5];   // (10, 442368)
    const float* fcb = (const float*)d_in[6];   // (10,)
    float* out = (float*)d_out;
    float* ws  = (float*)d_ws;

    const int B = 16;
    // workspace layout (floats)
    const long SZ_D  = 16L * 64 * 96 * 96;      // 9,437,184 (max distance buf)
    const long SZ_S  = 16L * 64 * 48 * 48;      // 2,359,296 (max sfm out)
    const long SZ_PN = 16L * 96 * 96;           //   147,456
    const long SZ_WN = 256;
    float* dbuf  = ws;
    float* sbuf  = dbuf + SZ_D;
    float* pn    = sbuf + SZ_S;
    float* wn    = pn + SZ_PN;
    float* feats = wn + SZ_WN;                  // (B, 3, 256, 576) = 7,077,888
    (void)ws_size; (void)in_sizes; (void)n_in; (void)out_size;

    for (int br = 0; br < 3; ++br) {
        // ---------------- Layer 1: Cin=1, 96x96, k=5, pad=2, Cout=64 ----------------
        {
            const int H = 96, W = 96, Kk = 5, pad = 2, Cin = 1, Cout = 64;
            const int Kdim = 25;
            const long HW = (long)H * W;
            const long M = (long)B * HW;                 // 147456
            const float* inb = x + (long)br * HW;        // channel br of (B,3,H,W)
            const long bstr = 3L * HW, cstr = HW;
            wnorm_kernel<<<(Cout + 63) / 64, 64, 0, stream>>>(
                w1 + (long)br * Cout * Kdim, Kdim, wn, Cout);
            pnorm_kernel<<<(int)((M + 255) / 256), 256, 0, stream>>>(
                inb, bstr, cstr, Cin, H, W, Kk, pad, pn, M);
            rbf_gemm_wmma<<<dim3((unsigned)(M / 32), Cout / 32), 32, 0, stream>>>(
                inb, bstr, cstr, Cin, H, W, Kk, pad, Kdim,
                w1 + (long)br * Cout * Kdim, pn, wn, dbuf, Cout);
            // crelu in place: k = 32, w = tri_w[br][0]
            crelu_kernel<2><<<(int)((M + 7) / 8), 256, 0, stream>>>(
                dbuf, dbuf, (long)Cout * HW, HW, Cout, (int)HW, B, 32, tw + br * 3 + 0);
            // sfm 96->48
            const long tot = (long)B * Cout * 48 * 48;   // 2,359,296
            sfm_kernel<<<(int)((tot + 255) / 256), 256, 0, stream>>>(dbuf, sbuf, H, W, tot);
        }
        // ---------------- Layer 2: Cin=64, 48x48, k=3, pad=1, Cout=128 ----------------
        {
            const int H = 48, W = 48, Kk = 3, pad = 1, Cin = 64, Cout = 128;
            const int Kdim = Cin * Kk * Kk;              // 576
            const long HW = (long)H * W;
            const long M = (long)B * HW;                 // 36864
            const long bstr = (long)Cin * HW, cstr = HW;
            wnorm_kernel<<<(Cout + 63) / 64, 64, 0, stream>>>(
                w2 + (long)br * Cout * Kdim, Kdim, wn, Cout);
            pnorm_kernel<<<(int)((M + 255) / 256), 256, 0, stream>>>(
                sbuf, bstr, cstr, Cin, H, W, Kk, pad, pn, M);
            rbf_gemm_wmma<<<dim3((unsigned)(M / 32), Cout / 32), 32, 0, stream>>>(
                sbuf, bstr, cstr, Cin, H, W, Kk, pad, Kdim,
                w2 + (long)br * Cout * Kdim, pn, wn, dbuf, Cout);
            crelu_kernel<4><<<(int)((M + 7) / 8), 256, 0, stream>>>(
                dbuf, dbuf, (long)Cout * HW, HW, Cout, (int)HW, B, 64, tw + br * 3 + 1);
            const long tot = (long)B * Cout * 24 * 24;   // 1,179,648
            sfm_kernel<<<(int)((tot + 255) / 256), 256, 0, stream>>>(dbuf, sbuf, H, W, tot);
        }
        // ---------------- Layer 3: Cin=128, 24x24, k=3, pad=1, Cout=256 ----------------
        {
            const int H = 24, W = 24, Kk = 3, pad = 1, Cin = 128, Cout = 256;
            const int Kdim = Cin * Kk * Kk;              // 1152
            const long HW = (long)H * W;
            const long M = (long)B * HW;                 // 9216
            const long bstr = (long)Cin * HW, cstr = HW;
            wnorm_kernel<<<(Cout + 63) / 64, 64, 0, stream>>>(
                w3 + (long)br * Cout * Kdim, Kdim, wn, Cout);
            pnorm_kernel<<<(int)((M + 255) / 256), 256, 0, stream>>>(
                sbuf, bstr, cstr, Cin, H, W, Kk, pad, pn, M);
            rbf_gemm_wmma<<<dim3((unsigned)(M / 32), Cout / 32), 32, 0, stream>>>(
                sbuf, bstr, cstr, Cin, H, W, Kk, pad, Kdim,
                w3 + (long)br * Cout * Kdim, pn, wn, dbuf, Cout);
            // crelu writes directly into feats (B, 3, 256, 576); k = int(0.6*256)=153
            float* ob = feats + (long)br * Cout * HW;
            crelu_kernel<8><<<(int)((M + 7) / 8), 256, 0, stream>>>(
                dbuf, ob, 3L * Cout * HW, HW, Cout, (int)HW, B, 153, tw + br * 3 + 2);
        }
    }
    // ---------------- FC: (16, 442368) x (10, 442368)^T + bias ----------------
    fc_kernel<<<dim3(16, 10), 256, 0, stream>>>(feats, fcw, fcb, out, 3 * 256 * 576);
}